// HadamardTransform_79353815761165
// MI455X (gfx1250) — compile-verified
//
#include <hip/hip_runtime.h>

// CDNA5 / gfx1250 FWHT via WMMA:  H4096 = H16 (x) H16 (x) H16  (Kronecker),
// applied as three in-LDS stages of 16x16x16 f32 matmuls using
// V_WMMA_F32_16X16X4_F32 (K=4, chained x4 for K=16). Exact ±1 coefficients,
// f32 accumulation -> matches the f32 reference up to summation order.

typedef __attribute__((ext_vector_type(2))) float v2f;
typedef __attribute__((ext_vector_type(8))) float v8f;

// One 16x16x16 matmul in place:  D = H16 * B, where
//   B[K][N] = lds[base + K*sK + N*sN],  D[M][N] = lds[base + M*sK + N*sN].
// WMMA f32 16x16x4 layouts (wave32):
//   A (16x4):  VGPR r, lanes 0-15: A[li][4c+r], lanes 16-31: A[li][4c+r+2]
//   B (4x16):  VGPR r, lanes 0-15: B[4c+r][li], lanes 16-31: B[4c+r+2][li]
//   D (16x16): VGPR r, lanes 0-15: D[r][li],    lanes 16-31: D[r+8][li]
__device__ __forceinline__ void hmul16_inplace(float* lds, int base, int sK, int sN,
                                               const v2f Hc[4], int li, int half) {
  const int nOff = base + li * sN;
  const int h2 = 2 * half;

  v2f B0, B1, B2, B3;
  B0.x = lds[nOff + (0  + h2) * sK];
  B0.y = lds[nOff + (1  + h2) * sK];
  B1.x = lds[nOff + (4  + h2) * sK];
  B1.y = lds[nOff + (5  + h2) * sK];
  B2.x = lds[nOff + (8  + h2) * sK];
  B2.y = lds[nOff + (9  + h2) * sK];
  B3.x = lds[nOff + (12 + h2) * sK];
  B3.y = lds[nOff + (13 + h2) * sK];

  v8f acc = {};
  // 8 args: (neg_a, A, neg_b, B, c_mod, C, reuse_a, reuse_b)
  acc = __builtin_amdgcn_wmma_f32_16x16x4_f32(false, Hc[0], false, B0, (short)0, acc, false, false);
  acc = __builtin_amdgcn_wmma_f32_16x16x4_f32(false, Hc[1], false, B1, (short)0, acc, false, false);
  acc = __builtin_amdgcn_wmma_f32_16x16x4_f32(false, Hc[2], false, B2, (short)0, acc, false, false);
  acc = __builtin_amdgcn_wmma_f32_16x16x4_f32(false, Hc[3], false, B3, (short)0, acc, false, false);

#pragma unroll
  for (int r = 0; r < 8; ++r)
    lds[nOff + (r + 8 * half) * sK] = acc[r];
}

__global__ __launch_bounds__(256) void
fwht4096_wmma_kernel(const float* __restrict__ x, const float* __restrict__ seed,
                     float* __restrict__ out, int nrows) {
  __shared__ __align__(16) float lds[4096];

  const int row  = blockIdx.x;
  const int t    = threadIdx.x;
  const int lane = t & 31;
  const int li   = lane & 15;
  const int half = lane >> 4;
  const int w    = t >> 5;      // wave id: 0..7

  const float4* __restrict__ x4 = (const float4*)(x + (size_t)row * 4096);
  const float4* __restrict__ s4 = (const float4*)seed;
  float4* __restrict__ o4       = (float4*)(out + (size_t)row * 4096);
  float4* l4                    = (float4*)lds;

  // Fold the 1/sqrt(4096) = 1/64 scale into the seeded load (2^-6, exact).
  const float scale = 0.015625f;
#pragma unroll
  for (int i = 0; i < 4; ++i) {
    const int idx = t + i * 256;          // float4 index, coalesced
    float4 v = x4[idx];
    float4 s = s4[idx];
    float4 r;
    r.x = v.x * s.x * scale;
    r.y = v.y * s.y * scale;
    r.z = v.z * s.z * scale;
    r.w = v.w * s.w * scale;
    l4[idx] = r;
  }

  // Prefetch next row while we compute this one (gfx1250 global_prefetch_b8).
  if (row + 1 < nrows)
    __builtin_prefetch((const void*)(x + (size_t)(row + 1) * 4096 + (size_t)t * 16), 0, 1);

  // Build H16 in the WMMA A-matrix register layout: H[m][k] = (-1)^popc(m&k).
  v2f Hc[4];
#pragma unroll
  for (int c = 0; c < 4; ++c) {
    const int k0 = 4 * c + 2 * half;
    Hc[c].x = (__popc(li & k0) & 1) ? -1.0f : 1.0f;
    Hc[c].y = (__popc(li & (k0 + 1)) & 1) ? -1.0f : 1.0f;
  }

  __syncthreads();

  // View row index as (hi,mid,lo), each 4 bits: addr = hi*256 + mid*16 + lo.
  // Stage 1: transform 'mid' axis.  B[K=mid][N=lo], per-hi slice (contiguous).
#pragma unroll
  for (int g = 0; g < 2; ++g) {
    const int G = 2 * w + g;
    hmul16_inplace(lds, G * 256, /*sK=*/16, /*sN=*/1, Hc, li, half);
  }
  __syncthreads();

  // Stage 2: transform 'lo' axis.  B[K=lo][N=(hi,mid) chunk].
#pragma unroll
  for (int g = 0; g < 2; ++g) {
    const int G = 2 * w + g;
    hmul16_inplace(lds, G * 256, /*sK=*/1, /*sN=*/16, Hc, li, half);
  }
  __syncthreads();

  // Stage 3: transform 'hi' axis.  B[K=hi][N=(mid,lo) chunk].
#pragma unroll
  for (int g = 0; g < 2; ++g) {
    const int G = 2 * w + g;
    hmul16_inplace(lds, G * 16, /*sK=*/256, /*sN=*/1, Hc, li, half);
  }
  __syncthreads();

#pragma unroll
  for (int i = 0; i < 4; ++i) {
    const int idx = t + i * 256;
    o4[idx] = l4[idx];
  }
}

extern "C" void kernel_launch(void* const* d_in, const int* in_sizes, int n_in,
                              void* d_out, int out_size, void* d_ws, size_t ws_size,
                              hipStream_t stream) {
  (void)n_in; (void)d_ws; (void)ws_size; (void)out_size;
  const float* x    = (const float*)d_in[0];
  const float* seed = (const float*)d_in[1];
  float* out        = (float*)d_out;

  const int n    = in_sizes[1];            // 4096 (row length)
  const int rows = in_sizes[0] / n;        // 8192

  fwht4096_wmma_kernel<<<rows, 256, 0, stream>>>(x, seed, out, rows);
}